// OctreeGroupNorm_74259984548360
// MI455X (gfx1250) — compile-verified
//
#include <hip/hip_runtime.h>

typedef float v2f __attribute__((ext_vector_type(2)));
typedef float v8f __attribute__((ext_vector_type(8)));

#define C_CH   64
#define NGROUP 8
#define CG     8
#define NBATCH 16
#define EPSF   1e-5f

// ---------------- Pass 0: segment boundaries (batch_id is sorted) ------------
__global__ void ogn_bounds_kernel(const int* __restrict__ bid, int N,
                                  int* __restrict__ starts) {
    int b = threadIdx.x;
    if (b > NBATCH) return;
    if (b == 0)      { starts[0] = 0;       return; }
    if (b == NBATCH) { starts[NBATCH] = N;  return; }
    int lo = 0, hi = N;                      // lower_bound: first idx with bid >= b
    while (lo < hi) {
        int mid = (lo + hi) >> 1;
        if (bid[mid] < b) lo = mid + 1; else hi = mid;
    }
    starts[b] = lo;
}

// ------- Pass A: per-(batch,channel) sum / sumsq via ones-matrix WMMA --------
// grid = (GB, NBATCH); each block owns one batch segment slice -> EXEC all-1s.
__global__ __launch_bounds__(256)
void ogn_stats_kernel(const float* __restrict__ data,
                      const int*   __restrict__ starts,
                      float*       __restrict__ partial) {
    const int b    = blockIdx.y;
    const int tid  = threadIdx.x;
    const int wave = tid >> 5;
    const int lane = tid & 31;
    const int half = lane >> 4;      // 0: rows r,r+1 ; 1: rows r+2,r+3
    const int n    = lane & 15;

    const int s0 = starts[b];
    const int s1 = starts[b + 1];
    const int nRows = s1 - s0;
    const int nFull = nRows >> 2;    // full 4-row chunks: no bounds checks

    const v2f onesA = {1.0f, 1.0f};  // A = 16x4 all-ones
    v8f accS[4] = {}, accQ[4] = {};

    const int  waveGlobal = blockIdx.x * 8 + wave;
    const int  W          = gridDim.x * 8;           // wave stride, in chunks
    const long step       = (long)W * 4 * C_CH;      // floats per stride

    // Lane-fixed base: row (s0 + 4*chunk + 2*half), column n.
    const float* p = data + ((long)s0 + 4L * waveGlobal + 2 * half) * C_CH + n;

    #pragma unroll 2
    for (int chunk = waveGlobal; chunk < nFull; chunk += W) {
        // 8 unconditional b32 loads, all immediate offsets off one address
        float x[2][4];
        #pragma unroll
        for (int e = 0; e < 2; ++e)
            #pragma unroll
            for (int cg = 0; cg < 4; ++cg)
                x[e][cg] = p[e * C_CH + cg * 16];
        #pragma unroll
        for (int cg = 0; cg < 4; ++cg) {
            v2f bS = {x[0][cg], x[1][cg]};
            v2f bQ = {x[0][cg] * x[0][cg], x[1][cg] * x[1][cg]};
            // D[m,n] = sum_k B[k,n] + C[m,n]  (column sums, accumulated in C)
            accS[cg] = __builtin_amdgcn_wmma_f32_16x16x4_f32(
                false, onesA, false, bS, (short)0, accS[cg], false, false);
            accQ[cg] = __builtin_amdgcn_wmma_f32_16x16x4_f32(
                false, onesA, false, bQ, (short)0, accQ[cg], false, false);
        }
        p += step;
    }

    // Tail chunk (<=3 rows), handled by exactly one wave; guard is wave-uniform
    // so EXEC stays all-1s (clamped loads + cndmask-to-zero, no branches).
    const int tailRows = nRows & 3;
    if (tailRows && waveGlobal == (nFull % W)) {
        const int r    = s0 + 4 * nFull;
        const int row0 = r + 2 * half;
        const int row1 = row0 + 1;
        const int c0   = row0 < s1 ? row0 : s1 - 1;   // clamp (s1>s0 here)
        const int c1   = row1 < s1 ? row1 : s1 - 1;
        const float* q0 = data + (long)c0 * C_CH + n;
        const float* q1 = data + (long)c1 * C_CH + n;
        #pragma unroll
        for (int cg = 0; cg < 4; ++cg) {
            float x0 = q0[cg * 16];
            float x1 = q1[cg * 16];
            x0 = (row0 < s1) ? x0 : 0.0f;
            x1 = (row1 < s1) ? x1 : 0.0f;
            v2f bS = {x0, x1};
            v2f bQ = {x0 * x0, x1 * x1};
            accS[cg] = __builtin_amdgcn_wmma_f32_16x16x4_f32(
                false, onesA, false, bS, (short)0, accS[cg], false, false);
            accQ[cg] = __builtin_amdgcn_wmma_f32_16x16x4_f32(
                false, onesA, false, bQ, (short)0, accQ[cg], false, false);
        }
    }

    // Every D row is identical (ones A): lanes 0-15, elem 0 hold channel sums.
    __shared__ float redS[8][C_CH];
    __shared__ float redQ[8][C_CH];
    if (lane < 16) {
        #pragma unroll
        for (int cg = 0; cg < 4; ++cg) {
            redS[wave][cg * 16 + lane] = accS[cg][0];
            redQ[wave][cg * 16 + lane] = accQ[cg][0];
        }
    }
    __syncthreads();
    if (tid < C_CH) {
        float s = 0.f, q = 0.f;
        #pragma unroll
        for (int w = 0; w < 8; ++w) { s += redS[w][tid]; q += redQ[w][tid]; }
        float* dst = partial + ((long)b * gridDim.x + blockIdx.x) * 128;
        dst[tid]      = s;
        dst[64 + tid] = q;
    }
}

// ---------------- Pass B: fold partials -> fused scale/shift tables ----------
__global__ __launch_bounds__(256)
void ogn_finalize_kernel(const int*   __restrict__ starts,
                         const float* __restrict__ partial, int gb,
                         const float* __restrict__ wgt,
                         const float* __restrict__ bias,
                         float* __restrict__ scaleT,
                         float* __restrict__ shiftT) {
    __shared__ float S[NBATCH][C_CH];
    __shared__ float Q[NBATCH][C_CH];
    __shared__ float meanG[NBATCH][NGROUP];
    __shared__ float isG[NBATCH][NGROUP];
    const int tid = threadIdx.x;

    for (int idx = tid; idx < NBATCH * C_CH; idx += 256) {
        const int b = idx >> 6, c = idx & 63;
        float s = 0.f, q = 0.f;
        const float* p = partial + (long)b * gb * 128;
        for (int i = 0; i < gb; ++i) {       // fixed order -> deterministic
            s += p[i * 128 + c];
            q += p[i * 128 + 64 + c];
        }
        S[b][c] = s; Q[b][c] = q;
    }
    __syncthreads();
    if (tid < NBATCH * NGROUP) {
        const int b = tid >> 3, g = tid & 7;
        float sg = 0.f, qg = 0.f;
        #pragma unroll
        for (int j = 0; j < CG; ++j) { sg += S[b][g * CG + j]; qg += Q[b][g * CG + j]; }
        const float cnt  = (float)(starts[b + 1] - starts[b]) * (float)CG;
        const float norm = 1.0f / (cnt + EPSF);
        const float m    = sg * norm;        // group mean (eps-perturbed, as ref)
        const float var  = (qg - 2.0f * m * sg + cnt * m * m) * norm;
        meanG[b][g] = m;
        isG[b][g]   = rsqrtf(var + EPSF);
    }
    __syncthreads();
    for (int idx = tid; idx < NBATCH * C_CH; idx += 256) {
        const int b = idx >> 6, c = idx & 63, g = c >> 3;
        const float sc = isG[b][g] * wgt[c];
        scaleT[idx] = sc;
        shiftT[idx] = bias[c] - meanG[b][g] * sc;
    }
}

// ---------------- Pass C: streaming apply, float4 ---------------------------
__global__ __launch_bounds__(256)
void ogn_apply_kernel(const float4* __restrict__ data4,
                      const int*    __restrict__ bid,
                      const float4* __restrict__ scaleT,
                      const float4* __restrict__ shiftT,
                      float4*       __restrict__ out4,
                      int total4) {
    int i = blockIdx.x * blockDim.x + threadIdx.x;
    const int stride = gridDim.x * blockDim.x;
    for (; i < total4; i += stride) {
        const int row = i >> 4;              // 16 float4 per row (C=64)
        const int q   = i & 15;
        const int b   = bid[row];            // L0-resident (shared cacheline)
        const float4 d  = data4[i];
        const float4 sc = scaleT[b * 16 + q];
        const float4 sh = shiftT[b * 16 + q];
        float4 o;
        o.x = fmaf(d.x, sc.x, sh.x);
        o.y = fmaf(d.y, sc.y, sh.y);
        o.z = fmaf(d.z, sc.z, sh.z);
        o.w = fmaf(d.w, sc.w, sh.w);
        out4[i] = o;
    }
}

extern "C" void kernel_launch(void* const* d_in, const int* in_sizes, int n_in,
                              void* d_out, int out_size, void* d_ws, size_t ws_size,
                              hipStream_t stream) {
    const float* data    = (const float*)d_in[0];
    const float* weights = (const float*)d_in[1];
    const float* bias    = (const float*)d_in[2];
    const int*   bid     = (const int*)d_in[3];
    // d_in[4] = batch_size scalar on device; reference fixes B=16.
    const int N = in_sizes[3];

    // ws layout: [0,256) starts | [256, +4KB) scale | [+4KB) shift | partials
    char*  ws      = (char*)d_ws;
    int*   starts  = (int*)ws;
    float* scaleT  = (float*)(ws + 256);
    float* shiftT  = (float*)(ws + 256 + NBATCH * C_CH * sizeof(float));
    size_t partOff = 256 + 2 * (size_t)NBATCH * C_CH * sizeof(float);
    partOff        = (partOff + 255) & ~(size_t)255;
    float* partial = (float*)(ws + partOff);

    // blocks per batch for the stats pass, clamped to available scratch
    int gb = 64;
    if (ws_size > partOff) {
        size_t avail = (ws_size - partOff) / ((size_t)NBATCH * 128 * sizeof(float));
        if ((size_t)gb > avail) gb = (int)avail;
    }
    if (gb < 1) gb = 1;

    ogn_bounds_kernel<<<1, 32, 0, stream>>>(bid, N, starts);

    dim3 sgrid(gb, NBATCH);
    ogn_stats_kernel<<<sgrid, 256, 0, stream>>>(data, starts, partial);

    ogn_finalize_kernel<<<1, 256, 0, stream>>>(starts, partial, gb,
                                               weights, bias, scaleT, shiftT);

    const int total4 = N * (C_CH / 4);
    int blocks = (total4 + 255) / 256;
    if (blocks > 12288) blocks = 12288;      // grid-stride
    ogn_apply_kernel<<<blocks, 256, 0, stream>>>((const float4*)data, bid,
                                                 (const float4*)scaleT,
                                                 (const float4*)shiftT,
                                                 (float4*)d_out, total4);
}